// VTransEFeaturePE_37434934952010
// MI455X (gfx1250) — compile-verified
//
#include <hip/hip_runtime.h>
#include <hip/hip_bf16.h>

typedef __attribute__((ext_vector_type(2))) float v2f;
typedef __attribute__((ext_vector_type(8))) float v8f;

#define NN  400
#define CC  151
#define DD  4096
#define EE  200
#define HH  512
#define PP  128
#define DEP 4424          // D + E + P (mult of 8)
#define MPAD 448          // NN padded to multiple of 64
#define KSM 152           // CC padded to multiple of 8 (softmax K)
#define NEPAD 224         // EE padded to multiple of 32
#define NOPAD 160         // CC padded to multiple of 32

// ---------------------------------------------------------------------------
// WMMA f32 GEMM, fully unpredicated loads:
//   A[Mpad x K]   (rows >= real M are zero-padded by producer, K % 8 == 0)
//   B[K x Npad]   (Npad % 32 == 0, grid.x == Npad/32)
//   C[.. x Cld]   stores guarded by (row < Mstore, col < Nstore)
// wave32: wave computes 64(M) x 32(N) = 4x2 tiles of v_wmma_f32_16x16x4_f32.
// block = 128 threads = 4 waves stacked in M.
// ---------------------------------------------------------------------------
__global__ __launch_bounds__(128) void gemm_wmma_f32(
    const float* __restrict__ A, const float* __restrict__ B,
    const float* __restrict__ bias, float* __restrict__ C,
    int Mpad, int Mstore, int Nstore, int K, int Npad, int Cld, int do_relu)
{
    const int lane = threadIdx.x & 31;
    const int wave = threadIdx.x >> 5;
    const int r    = lane & 15;   // row-in-tile (A) / col-in-tile (B,C)
    const int half = lane >> 4;   // K-pair select / M-half for C
    const int m0   = (blockIdx.y * 4 + wave) * 64;
    const int n0   = blockIdx.x * 32;
    if (m0 >= Mpad) return;       // wave-uniform exit (EXEC all-1 for WMMA)

    v8f acc[4][2];
    #pragma unroll
    for (int mt = 0; mt < 4; ++mt)
        #pragma unroll
        for (int nt = 0; nt < 2; ++nt)
            acc[mt][nt] = (v8f){0.f,0.f,0.f,0.f,0.f,0.f,0.f,0.f};

    // per-lane base pointers (8-byte aligned: K even, 2*half even, k0 mult 4)
    const float* Ap = A + (size_t)(m0 + r) * K + 2 * half;
    const float* Bp = B + (size_t)(2 * half) * Npad + n0 + r;

    #pragma unroll 2
    for (int k0 = 0; k0 < K; k0 += 4) {
        v2f a[4];
        #pragma unroll
        for (int mt = 0; mt < 4; ++mt)
            a[mt] = *(const v2f*)(Ap + (size_t)mt * 16 * K + k0);
        v2f b[2];
        #pragma unroll
        for (int nt = 0; nt < 2; ++nt) {
            const float* bp = Bp + (size_t)k0 * Npad + nt * 16;
            b[nt].x = bp[0];
            b[nt].y = bp[Npad];
        }
        #pragma unroll
        for (int mt = 0; mt < 4; ++mt)
            #pragma unroll
            for (int nt = 0; nt < 2; ++nt)
                acc[mt][nt] = __builtin_amdgcn_wmma_f32_16x16x4_f32(
                    false, a[mt], false, b[nt], (short)0, acc[mt][nt], false, false);
    }

    // C/D layout: VGPR v -> row = base + v + 8*half, col = base + (lane&15)
    #pragma unroll
    for (int nt = 0; nt < 2; ++nt) {
        const int col = n0 + nt * 16 + r;
        if (col >= Nstore) continue;
        const float bv = bias ? bias[col] : 0.f;
        #pragma unroll
        for (int mt = 0; mt < 4; ++mt) {
            #pragma unroll
            for (int v = 0; v < 8; ++v) {
                const int row = m0 + mt * 16 + half * 8 + v;
                if (row < Mstore) {
                    float val = acc[mt][nt][v] + bv;
                    if (do_relu) val = fmaxf(val, 0.f);
                    C[(size_t)row * Cld + col] = val;
                }
            }
        }
    }
}

// ---------------------------------------------------------------------------
// Row softmax into zero-padded (npad x cpad) buffer
// ---------------------------------------------------------------------------
__global__ void row_softmax_pad(const float* __restrict__ in, float* __restrict__ out,
                                int n, int c, int npad, int cpad)
{
    const int row = blockIdx.x * blockDim.x + threadIdx.x;
    if (row >= npad) return;
    float* o = out + (size_t)row * cpad;
    if (row >= n) {
        for (int j = 0; j < cpad; ++j) o[j] = 0.f;
        return;
    }
    const float* p = in + (size_t)row * c;
    float m = -INFINITY;
    for (int j = 0; j < c; ++j) m = fmaxf(m, p[j]);
    float s = 0.f;
    for (int j = 0; j < c; ++j) s += __expf(p[j] - m);
    const float inv = 1.f / s;
    for (int j = 0; j < c; ++j) o[j] = __expf(p[j] - m) * inv;
    for (int j = c; j < cpad; ++j) o[j] = 0.f;
}

// Generic zero-padded copy: src(rows x cols) -> dst(rpad x cpad)
__global__ void pad2d(const float* __restrict__ src, int rows, int cols,
                      float* __restrict__ dst, int rpad, int cpad)
{
    const size_t tot = (size_t)rpad * cpad;
    for (size_t i = (size_t)blockIdx.x * blockDim.x + threadIdx.x;
         i < tot; i += (size_t)gridDim.x * blockDim.x) {
        const int rr = (int)(i / cpad), cc = (int)(i % cpad);
        dst[i] = (rr < rows && cc < cols) ? src[(size_t)rr * cols + cc] : 0.f;
    }
}

// ---------------------------------------------------------------------------
// Position MLP: h = BN(pos@w1 + b1), pe = relu(h@w2 + b2)
// ---------------------------------------------------------------------------
__global__ void pos1_kernel(const float* __restrict__ pos, const float* __restrict__ w1,
                            const float* __restrict__ b1, const float* __restrict__ g,
                            const float* __restrict__ be, const float* __restrict__ mu,
                            const float* __restrict__ var, float* __restrict__ hbn)
{
    const int idx = blockIdx.x * blockDim.x + threadIdx.x;
    if (idx >= NN * 32) return;
    const int row = idx >> 5, j = idx & 31;
    const float* pi = pos + row * 9;
    float h = b1[j];
    #pragma unroll
    for (int t = 0; t < 9; ++t) h += pi[t] * w1[t * 32 + j];
    h = (h - mu[j]) * __frsqrt_rn(var[j] + 1e-5f) * g[j] + be[j];
    hbn[idx] = h;
}

__global__ void pos2_kernel(const float* __restrict__ hbn, const float* __restrict__ w2,
                            const float* __restrict__ b2, float* __restrict__ pe)
{
    const int idx = blockIdx.x * blockDim.x + threadIdx.x;
    if (idx >= NN * PP) return;
    const int row = idx / PP, o = idx % PP;
    const float* h = hbn + row * 32;
    float s = b2[o];
    #pragma unroll
    for (int t = 0; t < 32; ++t) s += h[t] * w2[t * PP + o];
    pe[idx] = fmaxf(s, 0.f);
}

// ---------------------------------------------------------------------------
// Concat builders (write MPAD x DEP, zero rows >= NN)
// ---------------------------------------------------------------------------
__global__ void build_rep1(const float* __restrict__ x, const float* __restrict__ oe,
                           const float* __restrict__ pe, float* __restrict__ rep)
{
    for (size_t i = (size_t)blockIdx.x * blockDim.x + threadIdx.x;
         i < (size_t)MPAD * DEP; i += (size_t)gridDim.x * blockDim.x) {
        const int row = (int)(i / DEP), c = (int)(i % DEP);
        float v = 0.f;
        if (row < NN) {
            if (c < DD)            v = x[(size_t)row * DD + c];
            else if (c < DD + EE)  v = oe[(size_t)row * EE + (c - DD)];
            else                   v = pe[(size_t)row * PP + (c - DD - EE)];
        }
        rep[i] = v;
    }
}

__global__ void build_rep2(const float* __restrict__ x, const float* __restrict__ pe,
                           const float* __restrict__ emb2_w, const int* __restrict__ preds,
                           float* __restrict__ rep, int* __restrict__ preds_out)
{
    const size_t t0 = (size_t)blockIdx.x * blockDim.x + threadIdx.x;
    if (t0 < NN) preds_out[t0] = preds[t0];   // export obj_preds (int32 bits)
    for (size_t i = t0; i < (size_t)MPAD * DEP; i += (size_t)gridDim.x * blockDim.x) {
        const int row = (int)(i / DEP), c = (int)(i % DEP);
        float v = 0.f;
        if (row < NN) {
            if (c < DD)            v = x[(size_t)row * DD + c];
            else if (c < DD + PP)  v = pe[(size_t)row * PP + (c - DD)];
            else                   v = emb2_w[(size_t)preds[row] * EE + (c - DD - PP)];
        }
        rep[i] = v;
    }
}

// ---------------------------------------------------------------------------
// Greedy NMS labeling: single block, 1024 threads (32 waves).
// IoU computed on the fly (never materialize the 400x400x151 overlap tensor).
// ---------------------------------------------------------------------------
__global__ __launch_bounds__(1024) void nms_kernel(
    const float* __restrict__ obj_dists, const float* __restrict__ boxes,
    float* __restrict__ scores, int* __restrict__ preds)
{
    const int tid = threadIdx.x;

    for (int j = tid; j < NN; j += 1024) preds[j] = 0;

    for (int row = tid; row < NN; row += 1024) {
        const float* p = obj_dists + (size_t)row * CC;
        float m = -INFINITY;
        for (int j = 0; j < CC; ++j) m = fmaxf(m, p[j]);
        float s = 0.f;
        for (int j = 0; j < CC; ++j) s += __expf(p[j] - m);
        const float inv = 1.f / s;
        float* o = scores + (size_t)row * CC;
        for (int j = 0; j < CC; ++j) o[j] = __expf(p[j] - m) * inv;
        o[0] = -1.0f;
    }
    __threadfence_block();
    __syncthreads();

    __shared__ float s_v[1024];
    __shared__ int   s_i[1024];
    __shared__ int   s_best;

    for (int it = 0; it < NN; ++it) {
        float bv = -INFINITY; int bidx = 0x7fffffff;
        for (int f = tid; f < NN * CC; f += 1024) {
            const float v = scores[f];
            if (v > bv) { bv = v; bidx = f; }
        }
        s_v[tid] = bv; s_i[tid] = bidx;
        __syncthreads();
        for (int sft = 512; sft > 0; sft >>= 1) {
            if (tid < sft) {
                const float ov = s_v[tid + sft]; const int oi = s_i[tid + sft];
                if (ov > s_v[tid] || (ov == s_v[tid] && oi < s_i[tid])) {
                    s_v[tid] = ov; s_i[tid] = oi;
                }
            }
            __syncthreads();
        }
        if (tid == 0) s_best = s_i[0];
        __syncthreads();

        const int flat = s_best;
        const int bi = flat / CC, ci = flat % CC;
        if (tid == 0) preds[bi] = ci;

        const float* pb = boxes + ((size_t)bi * CC + ci) * 4;
        const float px1 = pb[0], py1 = pb[1], px2 = pb[2], py2 = pb[3];
        const float parea = (px2 - px1 + 1.f) * (py2 - py1 + 1.f);

        for (int j = tid; j < NN; j += 1024) {
            const float* qb = boxes + ((size_t)j * CC + ci) * 4;
            const float qx1 = qb[0], qy1 = qb[1], qx2 = qb[2], qy2 = qb[3];
            const float ix = fminf(px2, qx2) - fmaxf(px1, qx1) + 1.f;
            const float iy = fminf(py2, qy2) - fmaxf(py1, qy1) + 1.f;
            const float inter = fmaxf(ix, 0.f) * fmaxf(iy, 0.f);
            const float qarea = (qx2 - qx1 + 1.f) * (qy2 - qy1 + 1.f);
            const float iou = inter / (parea + qarea - inter);
            if (iou >= 0.5f) scores[(size_t)j * CC + ci] = 0.0f;
        }
        __threadfence_block();
        __syncthreads();
        for (int c = tid; c < CC; c += 1024) scores[(size_t)bi * CC + c] = -1.0f;
        __threadfence_block();
        __syncthreads();
    }
}

// ---------------------------------------------------------------------------
extern "C" void kernel_launch(void* const* d_in, const int* in_sizes, int n_in,
                              void* d_out, int out_size, void* d_ws, size_t ws_size,
                              hipStream_t stream)
{
    (void)in_sizes; (void)n_in; (void)out_size; (void)ws_size;

    const float* x     = (const float*)d_in[0];
    const float* plog  = (const float*)d_in[1];
    const float* pos   = (const float*)d_in[2];
    const float* boxes = (const float*)d_in[3];
    const float* oe_w  = (const float*)d_in[4];
    const float* oe2_w = (const float*)d_in[5];
    const float* w1    = (const float*)d_in[6];
    const float* b1    = (const float*)d_in[7];
    const float* bg    = (const float*)d_in[8];
    const float* bb    = (const float*)d_in[9];
    const float* bm    = (const float*)d_in[10];
    const float* bvar  = (const float*)d_in[11];
    const float* w2    = (const float*)d_in[12];
    const float* b2    = (const float*)d_in[13];
    const float* wlin  = (const float*)d_in[14];
    const float* blin  = (const float*)d_in[15];
    const float* wout  = (const float*)d_in[16];
    const float* bout  = (const float*)d_in[17];
    const float* wfc   = (const float*)d_in[18];
    const float* bfc   = (const float*)d_in[19];

    float* out_dists = (float*)d_out;                        // 400*151
    int*   out_preds = (int*)((float*)d_out + NN * CC);      // 400 (int32 bits)
    float* out_edge  = (float*)d_out + NN * CC + NN;         // 400*512

    float* w = (float*)d_ws;
    float* p_sm   = w; w += MPAD * KSM;     // softmax(predict_logits), padded
    float* oew_p  = w; w += KSM * NEPAD;    // obj_embed_w padded (152x224)
    float* wout_p = w; w += HH * NOPAD;     // w_out padded (512x160)
    float* bout_p = w; w += NOPAD;          // b_out padded
    float* oe     = w; w += NN * EE;        // obj_embed (compact 400x200)
    float* hbn    = w; w += NN * 32;        // BN'd pos hidden
    float* pe     = w; w += NN * PP;        // pos_embed
    float* hidden = w; w += MPAD * HH;      // rep @ w_lin (padded rows kept)
    float* scores = w; w += NN * CC;        // NMS working scores
    int*   preds  = (int*)w; w += NN;       // NMS labels
    float* rep    = w;                      // MPAD*DEP, reused for obj_rel_rep

    // 0. pad small weight matrices / bias once per call
    pad2d<<<64, 256, 0, stream>>>(oe_w, CC, EE, oew_p, KSM, NEPAD);
    pad2d<<<128, 256, 0, stream>>>(wout, HH, CC, wout_p, HH, NOPAD);
    pad2d<<<1, 256, 0, stream>>>(bout, 1, CC, bout_p, 1, NOPAD);

    // 1. softmax(predict_logits) -> padded 448x152
    row_softmax_pad<<<(MPAD + 127) / 128, 128, 0, stream>>>(plog, p_sm, NN, CC, MPAD, KSM);

    // 2. pos MLP
    pos1_kernel<<<(NN * 32 + 255) / 256, 256, 0, stream>>>(pos, w1, b1, bg, bb, bm, bvar, hbn);
    pos2_kernel<<<(NN * PP + 255) / 256, 256, 0, stream>>>(hbn, w2, b2, pe);

    // 3. obj_embed = softmax @ obj_embed_w   (448x152 @ 152x224 -> 400x200)
    {
        dim3 g(NEPAD / 32, MPAD / 256 + (MPAD % 256 ? 1 : 0));
        gemm_wmma_f32<<<g, 128, 0, stream>>>(p_sm, oew_p, nullptr, oe,
                                             MPAD, NN, EE, KSM, NEPAD, EE, 0);
    }
    // 4. rep = [x | obj_embed | pos_embed], zero-padded to 448 rows
    build_rep1<<<512, 256, 0, stream>>>(x, oe, pe, rep);

    // 5. hidden = rep @ w_lin + b_lin        (448x4424 @ 4424x512 -> 448x512)
    {
        dim3 g(HH / 32, MPAD / 256 + (MPAD % 256 ? 1 : 0));
        gemm_wmma_f32<<<g, 128, 0, stream>>>(rep, wlin, blin, hidden,
                                             MPAD, MPAD, HH, DEP, HH, HH, 0);
    }
    // 6. obj_dists = hidden @ w_out + b_out  (448x512 @ 512x160 -> 400x151) -> d_out
    {
        dim3 g(NOPAD / 32, MPAD / 256 + (MPAD % 256 ? 1 : 0));
        gemm_wmma_f32<<<g, 128, 0, stream>>>(hidden, wout_p, bout_p, out_dists,
                                             MPAD, NN, CC, HH, NOPAD, CC, 0);
    }
    // 7. greedy NMS labels (serial loop, single block, on-the-fly IoU)
    nms_kernel<<<1, 1024, 0, stream>>>(out_dists, boxes, scores, preds);

    // 8. obj_rel_rep = [x | pos_embed | obj_embed2_w[preds]]; export preds
    build_rep2<<<512, 256, 0, stream>>>(x, pe, oe2_w, preds, rep, out_preds);

    // 9. edge_ctx = relu(obj_rel_rep @ w_fc + b_fc) -> d_out
    {
        dim3 g(HH / 32, MPAD / 256 + (MPAD % 256 ? 1 : 0));
        gemm_wmma_f32<<<g, 128, 0, stream>>>(rep, wfc, bfc, out_edge,
                                             MPAD, NN, HH, DEP, HH, HH, 1);
    }
}